// Adaptive_pooling_90013924589651
// MI455X (gfx1250) — compile-verified
//
#include <hip/hip_runtime.h>
#include <math.h>

#define N_NODES 2048
#define DEG     32
#define NE      (N_NODES * DEG)   // 65536 edges
#define DIM     512
#define NN      ((size_t)N_NODES * (size_t)N_NODES)

typedef __attribute__((ext_vector_type(2))) float v2f;
typedef __attribute__((ext_vector_type(8))) float v8f;

// ---------------------------------------------------------------------------
// 1) Row inverse norms: rnorm[n] = 1 / max(||emb[n]||, 1e-12)
// ---------------------------------------------------------------------------
__global__ __launch_bounds__(256)
void rownorm_kernel(const float* __restrict__ emb, float* __restrict__ rnorm) {
    __shared__ float red[256];
    const int row = blockIdx.x;
    const int t = threadIdx.x;
    const float* x = emb + (size_t)row * DIM;
    float a0 = x[t], a1 = x[t + 256];
    red[t] = a0 * a0 + a1 * a1;
    __syncthreads();
    for (int s = 128; s > 0; s >>= 1) {
        if (t < s) red[t] += red[t + s];
        __syncthreads();
    }
    if (t == 0) rnorm[row] = 1.0f / fmaxf(sqrtf(red[0]), 1e-12f);
}

// ---------------------------------------------------------------------------
// 2) Per-edge cosine similarity: one wave32 per edge
// ---------------------------------------------------------------------------
__global__ __launch_bounds__(256)
void edge_lin_kernel(const float* __restrict__ emb,
                     const int* __restrict__ src, const int* __restrict__ dst,
                     const float* __restrict__ rnorm, float* __restrict__ lin) {
    const int wave = (int)((blockIdx.x * blockDim.x + threadIdx.x) >> 5);
    const int lane = threadIdx.x & 31;
    if (wave >= NE) return;
    const int s = src[wave], d = dst[wave];
    const float* xs = emb + (size_t)s * DIM;
    const float* xd = emb + (size_t)d * DIM;
    float acc = 0.f;
    #pragma unroll 4
    for (int i = lane; i < DIM; i += 32) acc = fmaf(xs[i], xd[i], acc);
    for (int off = 16; off; off >>= 1) acc += __shfl_xor(acc, off, 32);
    if (lane == 0) lin[wave] = acc * rnorm[s] * rnorm[d];
}

// ---------------------------------------------------------------------------
// 3) Segment softmax over src: DEG==32 edges per src -> one wave per segment
// ---------------------------------------------------------------------------
__global__ __launch_bounds__(256)
void softmax_kernel(const float* __restrict__ lin, float* __restrict__ fitness) {
    const int s = (int)((blockIdx.x * blockDim.x + threadIdx.x) >> 5);
    const int lane = threadIdx.x & 31;
    if (s >= N_NODES) return;
    float v = lin[s * DEG + lane];
    float m = v;
    for (int off = 16; off; off >>= 1) m = fmaxf(m, __shfl_xor(m, off, 32));
    float ex = expf(v - m);
    float sum = ex;
    for (int off = 16; off; off >>= 1) sum += __shfl_xor(sum, off, 32);
    fitness[s * DEG + lane] = ex / sum;
}

// ---------------------------------------------------------------------------
// 4) scores = segment_mean(fitness over dst)
// ---------------------------------------------------------------------------
__global__ __launch_bounds__(256)
void scores_scatter_kernel(const int* __restrict__ dst, const float* __restrict__ fit,
                           float* __restrict__ sumf, float* __restrict__ cnt) {
    const int e = blockIdx.x * 256 + threadIdx.x;
    if (e >= NE) return;
    const int d = dst[e];
    atomicAdd(&sumf[d], fit[e]);
    atomicAdd(&cnt[d], 1.0f);
}

__global__ __launch_bounds__(256)
void scores_final_kernel(const float* __restrict__ sumf, const float* __restrict__ cnt,
                         float* __restrict__ scores) {
    const int n = blockIdx.x * 256 + threadIdx.x;
    if (n >= N_NODES) return;
    scores[n] = sumf[n] / fmaxf(cnt[n], 1.0f);
}

// ---------------------------------------------------------------------------
// 5) del_mask / has_src / cluster_mask
// ---------------------------------------------------------------------------
__global__ __launch_bounds__(256)
void bad_scatter_kernel(const int* __restrict__ src, const int* __restrict__ dst,
                        const float* __restrict__ scores,
                        int* __restrict__ del, int* __restrict__ hassrc) {
    const int e = blockIdx.x * 256 + threadIdx.x;
    if (e >= NE) return;
    const int s = src[e], d = dst[e];
    atomicOr(&hassrc[s], 1);
    if (scores[s] - scores[d] < 0.f) atomicOr(&del[s], 1);
}

__global__ __launch_bounds__(256)
void cluster_kernel(const int* __restrict__ hassrc, const int* __restrict__ del,
                    int* __restrict__ cluster) {
    const int n = blockIdx.x * 256 + threadIdx.x;
    if (n >= N_NODES) return;
    cluster[n] = (hassrc[n] && !del[n]) ? 1 : 0;
}

// ---------------------------------------------------------------------------
// 6) Scatter edges into dense S (d_out) and dense A (ws); in_node flags
//    Edges are unique (src,dst) pairs w/o self loops -> plain stores
// ---------------------------------------------------------------------------
__global__ __launch_bounds__(256)
void edge_sa_kernel(const int* __restrict__ src, const int* __restrict__ dst,
                    const float* __restrict__ fit, const int* __restrict__ cluster,
                    int* __restrict__ innode,
                    float* __restrict__ S, float* __restrict__ A) {
    const int e = blockIdx.x * 256 + threadIdx.x;
    if (e >= NE) return;
    const int s = src[e], d = dst[e];
    const int mi = cluster[d];
    if (mi) {
        atomicOr(&innode[s], 1);
        atomicOr(&innode[d], 1);
        S[(size_t)s * N_NODES + d] = fit[e];
    }
    A[(size_t)s * N_NODES + d] = 1.0f;
}

__global__ __launch_bounds__(256)
void node_finalize_kernel(const int* __restrict__ cluster, const int* __restrict__ innode,
                          float* __restrict__ S, int* __restrict__ colm_i,
                          float* __restrict__ colm_out) {
    const int n = blockIdx.x * 256 + threadIdx.x;
    if (n >= N_NODES) return;
    const int cm = cluster[n];
    const int ni = innode[n] ? 0 : 1;               // non_in
    S[(size_t)n * N_NODES + n] = (float)cm + (float)ni;  // diagv
    const int col = cm | ni;                        // col_mask
    colm_i[n] = col;
    colm_out[n] = col ? 1.0f : 0.0f;
}

// ---------------------------------------------------------------------------
// 7) WMMA f32 GEMM: C[M,Nc] = Aop[M,K] * B[K,Nc] using V_WMMA_F32_16X16X4_F32
//    Register-blocked: each wave owns a 32x64 output tile (2 M-subtiles x
//    4 N-subtiles = 8 v8f accumulators). Per k-step: 2 A-frag loads +
//    4 B-frag loads feed 8 wmma ops (~1.25 vmem/wmma vs 3 unblocked).
//    TRANSA: Aop(m,k) = A[k*M + m]   (A stored K-major, lda == M)
//    BINA:   Aop(m,k) = (val > 0) ? 1 : 0
//    16x4 f32 A fragment: lanes 0-15 hold K=k0,k0+1 ; lanes 16-31 K=k0+2,k0+3
//    f32 C fragment: VGPR r -> row = r + 8*(lane>=16), col = lane%16
// ---------------------------------------------------------------------------
template <int TRANSA, int BINA>
__global__ __launch_bounds__(128)
void gemm_wmma_f32(const float* __restrict__ A, const float* __restrict__ B,
                   float* __restrict__ C, int M, int Nc, int K) {
    const int lane = threadIdx.x & 31;
    const int wave = threadIdx.x >> 5;
    const int row0 = (blockIdx.y * 4 + wave) * 32;   // 32 rows per wave
    const int col0 = blockIdx.x * 64;                // 64 cols per wave
    const int half = lane >> 4;                      // 0 or 1
    const int l    = lane & 15;

    v8f acc[2][4] = {};

    #pragma unroll 2
    for (int k0 = 0; k0 < K; k0 += 4) {
        const int ka = k0 + 2 * half;

        v2f a[2];
        #pragma unroll
        for (int i = 0; i < 2; ++i) {
            const int m = row0 + 16 * i + l;
            if (TRANSA) {
                float a0 = A[(size_t)ka * M + m];
                float a1 = A[(size_t)(ka + 1) * M + m];
                if (BINA) {
                    a0 = (a0 > 0.f) ? 1.f : 0.f;
                    a1 = (a1 > 0.f) ? 1.f : 0.f;
                }
                a[i].x = a0; a[i].y = a1;
            } else {
                a[i] = *(const v2f*)(A + (size_t)m * K + ka);  // 8B aligned
            }
        }

        v2f b[4];
        #pragma unroll
        for (int j = 0; j < 4; ++j) {
            const int n = col0 + 16 * j + l;
            b[j].x = B[(size_t)ka * Nc + n];
            b[j].y = B[(size_t)(ka + 1) * Nc + n];
            // prefetch the B rows 2 k-steps (8 k) ahead (global_prefetch_b8)
            __builtin_prefetch(B + (size_t)(ka + 8) * Nc + n, 0, 1);
        }

        #pragma unroll
        for (int i = 0; i < 2; ++i)
            #pragma unroll
            for (int j = 0; j < 4; ++j)
                acc[i][j] = __builtin_amdgcn_wmma_f32_16x16x4_f32(
                    /*neg_a=*/false, a[i], /*neg_b=*/false, b[j],
                    /*c_mod=*/(short)0, acc[i][j],
                    /*reuse_a=*/false, /*reuse_b=*/false);
    }

    #pragma unroll
    for (int i = 0; i < 2; ++i)
        #pragma unroll
        for (int j = 0; j < 4; ++j)
            #pragma unroll
            for (int r = 0; r < 8; ++r)
                C[(size_t)(row0 + 16 * i + r + 8 * half) * Nc + (col0 + 16 * j + l)] =
                    acc[i][j][r];
}

// ---------------------------------------------------------------------------
// 8) Emat diagonal fixup: where diag==0 && col_mask -> 1.0
// ---------------------------------------------------------------------------
__global__ __launch_bounds__(256)
void diagfix_kernel(float* __restrict__ Emat, const int* __restrict__ colm) {
    const int n = blockIdx.x * 256 + threadIdx.x;
    if (n >= N_NODES) return;
    const size_t idx = (size_t)n * N_NODES + n;
    const float v = Emat[idx];
    if (v == 0.f && colm[n]) Emat[idx] = 1.0f;
}

// ---------------------------------------------------------------------------
// 9) Column nonzero counts of S: cnt[i] = sum_k (S[k][i] > 0)
// ---------------------------------------------------------------------------
__global__ __launch_bounds__(256)
void colcnt_kernel(const float* __restrict__ S, float* __restrict__ cntcol) {
    __shared__ float red[256];
    const int col = blockIdx.x, t = threadIdx.x;
    float c = 0.f;
    for (int k = t; k < N_NODES; k += 256)
        c += (S[(size_t)k * N_NODES + col] > 0.f) ? 1.f : 0.f;
    red[t] = c;
    __syncthreads();
    for (int s = 128; s > 0; s >>= 1) {
        if (t < s) red[t] += red[t + s];
        __syncthreads();
    }
    if (t == 0) cntcol[col] = red[0];
}

// ---------------------------------------------------------------------------
// 10) pooled scale + col_mask: pooled[i,:] = colm ? raw / max(cnt,1) : 0
// ---------------------------------------------------------------------------
__global__ __launch_bounds__(256)
void pooled_scale_kernel(float* __restrict__ pooled, const float* __restrict__ cntcol,
                         const int* __restrict__ colm) {
    const int idx = blockIdx.x * 256 + threadIdx.x;
    if (idx >= N_NODES * DIM) return;
    const int i = idx / DIM;
    const float v = pooled[idx] / fmaxf(cntcol[i], 1.0f);
    pooled[idx] = colm[i] ? v : 0.0f;
}

// ---------------------------------------------------------------------------
extern "C" void kernel_launch(void* const* d_in, const int* in_sizes, int n_in,
                              void* d_out, int out_size, void* d_ws, size_t ws_size,
                              hipStream_t stream) {
    (void)in_sizes; (void)n_in; (void)out_size; (void)ws_size;

    const float* emb  = (const float*)d_in[0];
    const int*   eidx = (const int*)d_in[1];
    const int*   src  = eidx;
    const int*   dst  = eidx + NE;

    // d_out: pooled[N,D] | Emat[N,N] | S[N,N] | fitness[E] | col_mask[N]
    float* out        = (float*)d_out;
    float* pooled     = out;
    float* Emat       = pooled + (size_t)N_NODES * DIM;
    float* S          = Emat + NN;
    float* fitness    = S + NN;
    float* colm_out   = fitness + NE;

    // workspace layout
    char* ws = (char*)d_ws;
    size_t off = 0;
    auto alloc = [&](size_t bytes) { void* p = ws + off; off = (off + bytes + 255) & ~(size_t)255; return p; };
    float* rnorm   = (float*)alloc(N_NODES * 4);
    float* lin     = (float*)alloc(NE * 4);
    float* sumf    = (float*)alloc(N_NODES * 4);   // zeroed
    float* cntin   = (float*)alloc(N_NODES * 4);   // zeroed
    float* scores  = (float*)alloc(N_NODES * 4);
    int*   del     = (int*)alloc(N_NODES * 4);     // zeroed
    int*   hassrc  = (int*)alloc(N_NODES * 4);     // zeroed
    int*   cluster = (int*)alloc(N_NODES * 4);
    int*   innode  = (int*)alloc(N_NODES * 4);     // zeroed
    int*   colm_i  = (int*)alloc(N_NODES * 4);
    float* cntcol  = (float*)alloc(N_NODES * 4);
    float* Adense  = (float*)alloc(NN * 4);        // zeroed
    float* Bdense  = (float*)alloc(NN * 4);        // fully overwritten by GEMM1

    // zero the accumulator/flag region (sumf .. cntcol) in one shot, plus A and S
    hipMemsetAsync(sumf, 0, (size_t)((char*)Adense - (char*)sumf), stream);
    hipMemsetAsync(Adense, 0, NN * 4, stream);
    hipMemsetAsync(S, 0, NN * 4, stream);

    // pipeline
    rownorm_kernel<<<N_NODES, 256, 0, stream>>>(emb, rnorm);
    edge_lin_kernel<<<NE / 8, 256, 0, stream>>>(emb, src, dst, rnorm, lin);
    softmax_kernel<<<N_NODES / 8, 256, 0, stream>>>(lin, fitness);
    scores_scatter_kernel<<<NE / 256, 256, 0, stream>>>(dst, fitness, sumf, cntin);
    scores_final_kernel<<<N_NODES / 256, 256, 0, stream>>>(sumf, cntin, scores);
    bad_scatter_kernel<<<NE / 256, 256, 0, stream>>>(src, dst, scores, del, hassrc);
    cluster_kernel<<<N_NODES / 256, 256, 0, stream>>>(hassrc, del, cluster);
    edge_sa_kernel<<<NE / 256, 256, 0, stream>>>(src, dst, fitness, cluster, innode, S, Adense);
    node_finalize_kernel<<<N_NODES / 256, 256, 0, stream>>>(cluster, innode, S, colm_i, colm_out);

    // dense WMMA chain (all f32, L2-resident operands)
    // each block: 4 waves x (32x64) = 128x64 output tile
    dim3 blk(128);
    dim3 grdNN(N_NODES / 64, N_NODES / 128);
    // B = A @ S
    gemm_wmma_f32<0, 0><<<grdNN, blk, 0, stream>>>(Adense, S, Bdense, N_NODES, N_NODES, N_NODES);
    // Emat = S^T @ B
    gemm_wmma_f32<1, 0><<<grdNN, blk, 0, stream>>>(S, Bdense, Emat, N_NODES, N_NODES, N_NODES);
    diagfix_kernel<<<N_NODES / 256, 256, 0, stream>>>(Emat, colm_i);

    // pooled = (S>0)^T @ emb, then scale by 1/max(cnt,1) and mask
    colcnt_kernel<<<N_NODES, 256, 0, stream>>>(S, cntcol);
    dim3 grdP(DIM / 64, N_NODES / 128);
    gemm_wmma_f32<1, 1><<<grdP, blk, 0, stream>>>(S, emb, pooled, N_NODES, DIM, N_NODES);
    pooled_scale_kernel<<<(N_NODES * DIM) / 256, 256, 0, stream>>>(pooled, cntcol, colm_i);
}